// AttentionLayer_39659728011615
// MI455X (gfx1250) — compile-verified
//
#include <hip/hip_runtime.h>

// Problem constants from the reference
#define NB   4
#define TDIM 512   // TD
#define EDIM 512   // TE
#define DDIM 128   // D

typedef float v2f __attribute__((ext_vector_type(2)));
typedef float v8f __attribute__((ext_vector_type(8)));

// LDS strides chosen for bank-conflict-free access (64 banks x 4B):
//  - DEC_LD = 132 (≡4 mod 64): GEMM A-frag reads, lanes indexed by m (0..15): banks 4m+k distinct
//  - ENC_LD = 129 (≡1 mod 64): score phase, lanes indexed by e (32 consecutive): banks e+d distinct
//  - SC_LD  = 516 (≡4 mod 64): GEMM2 A-frag reads, same argument as DEC_LD
#define DEC_LD 132
#define ENC_LD 129
#define SC_LD  (EDIM + 4)
#define ECHUNK 64
#define NCHUNK (EDIM / ECHUNK)

// ---- CDNA5 feature probes (host pass sees neither; device pass picks them up) ----
#if __has_builtin(__builtin_amdgcn_global_load_async_to_lds_b32) && \
    __has_builtin(__builtin_amdgcn_s_wait_asynccnt)
#define USE_ASYNC 1
#else
#define USE_ASYNC 0
#endif

__device__ __forceinline__ float fast_tanh(float x) {
#if __has_builtin(__builtin_amdgcn_tanhf)
    return __builtin_amdgcn_tanhf(x);            // v_tanh_f32: 1 trans op
#else
    // tanh(x) = 1 - 2/(1 + e^{2x}); exact +-1 saturation at the infinities
    const float TWO_LOG2E = 2.885390081777926774f;
    float e = __builtin_amdgcn_exp2f(x * TWO_LOG2E);
    float r = __builtin_amdgcn_rcpf(1.0f + e);
    return __builtin_fmaf(-2.0f, r, 1.0f);
#endif
}

#if USE_ASYNC
// Builtin signature (per hipcc diagnostic): (addrspace(1) int*, addrspace(3) int*, imm, imm)
typedef __attribute__((address_space(1))) int as1_int;
typedef __attribute__((address_space(3))) int as3_int;
#endif

// Stage one 64x128 enc chunk into LDS (stride-129 rows).
__device__ __forceinline__ void issue_enc_chunk(const float* __restrict__ gsrc,
                                                float* __restrict__ ldst, int tid) {
#pragma unroll
    for (int j = 0; j < (ECHUNK * DDIM) / 256; ++j) {   // 32 elements / thread
        const int i = tid + 256 * j;
        const float* s = gsrc + i;
        float*       d = ldst + (i >> 7) * ENC_LD + (i & 127);
#if USE_ASYNC
        __builtin_amdgcn_global_load_async_to_lds_b32((as1_int*)s, (as3_int*)d, 0, 0);
#else
        *d = *s;
#endif
    }
}

__launch_bounds__(256)
__global__ void additive_attn_kernel(const float* __restrict__ dec,   // [B, TD, D]
                                     const float* __restrict__ enc,   // [B, TE, D]
                                     const float* __restrict__ W,     // [D, D]
                                     const float* __restrict__ V,     // [D]
                                     float* __restrict__ out)         // [B, TD, D]
{
    __shared__ float sDec [16 * DEC_LD];        // dec tile          (8.4 KB)
    __shared__ float sDecT[16 * DEC_LD];        // dec @ W tile      (8.4 KB)
    __shared__ float sEnc [2][ECHUNK * ENC_LD]; // enc double-buffer (66 KB)
    __shared__ float sScore[16 * SC_LD];        // scores/attn       (33 KB)
    __shared__ float sV[DDIM];

    const int tid  = threadIdx.x;
    const int lane = tid & 31;
    const int wave = tid >> 5;               // 0..7
    const int blk  = blockIdx.x;             // 0..127
    const int b    = blk >> 5;
    const int t0   = (blk & 31) << 4;

    const float* decB = dec + ((size_t)b * TDIM + t0) * DDIM;
    const float* encB = enc + (size_t)b * EDIM * DDIM;

#if USE_ASYNC
    // Chunk 0 streams into LDS while GEMM1 runs (tracked by ASYNCcnt).
    issue_enc_chunk(encB, sEnc[0], tid);
#endif

    if (tid < DDIM) sV[tid] = V[tid];
    // 16x128 dec rows are contiguous in memory -> fully coalesced copy
    for (int i = tid; i < 16 * DDIM; i += 256)
        sDec[(i >> 7) * DEC_LD + (i & 127)] = decB[i];
    __syncthreads();

    const int m    = lane & 15;
    const int hi   = lane >> 4;       // 0: K={0,1}/M=r ; 1: K={2,3}/M=r+8
    const int koff = hi << 1;
    const int n    = (wave << 4) + m; // this wave's N-tile column

    // ---------------- GEMM1: DecT[16x128] = Dec[16x128] x W[128x128] ----------------
    {
        v8f c = {};
        for (int k0 = 0; k0 < DDIM; k0 += 4) {
            v2f a, bm;
            a.x  = sDec[m * DEC_LD + k0 + koff];
            a.y  = sDec[m * DEC_LD + k0 + koff + 1];
            bm.x = W[(k0 + koff)     * DDIM + n];
            bm.y = W[(k0 + koff + 1) * DDIM + n];
            c = __builtin_amdgcn_wmma_f32_16x16x4_f32(false, a, false, bm,
                                                      (short)0, c, false, false);
        }
#pragma unroll
        for (int r = 0; r < 8; ++r)
            sDecT[(r + (hi << 3)) * DEC_LD + n] = c[r];
    }
    __syncthreads();

    // ---------------- Score phase (dominant cost: 1M v_tanh_f32 per WG) ----------------
    // score[t][e] = sum_d tanh(DecT[t][d] + Enc[e][d]) * V[d]
    for (int c = 0; c < NCHUNK; ++c) {
        const int cur = c & 1;
#if USE_ASYNC
        __builtin_amdgcn_s_wait_asynccnt(0);  // this thread's chunk-c writes landed
        __syncthreads();                      // everyone's landed; buf[cur^1] fully read
        if (c + 1 < NCHUNK)                   // prefetch next chunk into other buffer
            issue_enc_chunk(encB + (c + 1) * ECHUNK * DDIM, sEnc[cur ^ 1], tid);
#else
        __syncthreads();                      // buf[cur] fully read (2 iters ago)
        issue_enc_chunk(encB + c * ECHUNK * DDIM, sEnc[cur], tid);
        __syncthreads();
#endif
        for (int ent = tid; ent < 16 * ECHUNK; ent += 256) { // exactly 4 iters/thread
            const int e = ent & (ECHUNK - 1);
            const int t = ent >> 6;
            const float* dt = &sDecT[t * DEC_LD];    // wave-uniform broadcast
            const float* er = &sEnc[cur][e * ENC_LD];// conflict-free (stride 129)
            float s = 0.f;
#pragma unroll 8
            for (int d = 0; d < DDIM; ++d)
                s = __builtin_fmaf(fast_tanh(dt[d] + er[d]), sV[d], s);
            sScore[t * SC_LD + c * ECHUNK + e] = s;
        }
#if !USE_ASYNC
        __syncthreads();
#endif
    }
#if USE_ASYNC
    __syncthreads();   // all scores written
#endif

    // ---------------- Softmax over e, one wave32 per 2 rows ----------------
    const float LOG2E = 1.442695040888963407f;
    for (int row = wave; row < 16; row += 8) {
        float* sr = &sScore[row * SC_LD];
        float mx = -__builtin_inff();
        for (int e = lane; e < EDIM; e += 32) mx = fmaxf(mx, sr[e]);
#pragma unroll
        for (int off = 16; off > 0; off >>= 1) mx = fmaxf(mx, __shfl_xor(mx, off, 32));
        float sum = 0.f;
        for (int e = lane; e < EDIM; e += 32) {
            float p = __builtin_amdgcn_exp2f((sr[e] - mx) * LOG2E);
            sr[e] = p;
            sum += p;
        }
#pragma unroll
        for (int off = 16; off > 0; off >>= 1) sum += __shfl_xor(sum, off, 32);
        float inv = __builtin_amdgcn_rcpf(sum);
        for (int e = lane; e < EDIM; e += 32) sr[e] *= inv;
    }
    __syncthreads();

    // ---------------- GEMM2: Ctx[16x128] = Attn[16x512] x Enc[512x128] ----------------
    {
        v8f c = {};
        for (int k0 = 0; k0 < EDIM; k0 += 4) {
            v2f a, bm;
            a.x  = sScore[m * SC_LD + k0 + koff];
            a.y  = sScore[m * SC_LD + k0 + koff + 1];
            bm.x = encB[(k0 + koff)     * DDIM + n]; // L2-resident (enc[b] = 256 KB)
            bm.y = encB[(k0 + koff + 1) * DDIM + n];
            c = __builtin_amdgcn_wmma_f32_16x16x4_f32(false, a, false, bm,
                                                      (short)0, c, false, false);
        }
        float* outB = out + ((size_t)b * TDIM + t0) * DDIM;
#pragma unroll
        for (int r = 0; r < 8; ++r)
            outB[(r + (hi << 3)) * DDIM + n] = c[r];
    }
}

extern "C" void kernel_launch(void* const* d_in, const int* in_sizes, int n_in,
                              void* d_out, int out_size, void* d_ws, size_t ws_size,
                              hipStream_t stream) {
    (void)in_sizes; (void)n_in; (void)d_ws; (void)ws_size; (void)out_size;
    const float* dec = (const float*)d_in[0]; // [4,512,128]
    const float* enc = (const float*)d_in[1]; // [4,512,128]
    const float* W   = (const float*)d_in[2]; // [128,128]
    const float* V   = (const float*)d_in[3]; // [128,1]
    float* out       = (float*)d_out;         // [4,512,128]

    dim3 grid(NB * (TDIM / 16));  // 128 workgroups
    dim3 block(256);              // 8 wave32s
    additive_attn_kernel<<<grid, block, 0, stream>>>(dec, enc, W, V, out);
}